// Attention_8839042695176
// MI455X (gfx1250) — compile-verified
//
#include <hip/hip_runtime.h>
#include <cmath>

typedef float v2f __attribute__((ext_vector_type(2)));
typedef float v8f __attribute__((ext_vector_type(8)));

#define HW     2304      // 48*48
#define NTILE  144       // 2304/16
#define SCALEF 0.35355339059327373f

static __device__ __forceinline__ v8f wmma4(v2f a, v2f b, v8f c) {
  // V_WMMA_F32_16X16X4_F32 : D = A(16x4) * B(4x16) + C(16x16), all f32
  return __builtin_amdgcn_wmma_f32_16x16x4_f32(false, a, false, b, (short)0, c,
                                               false, false);
}

// ---------------- 1x1 conv -> q, k, vT ----------------
__global__ void k_qkv(const float* __restrict__ f, const float* __restrict__ wqkv,
                      float* __restrict__ q, float* __restrict__ k,
                      float* __restrict__ vT) {
  int t = blockIdx.x * blockDim.x + threadIdx.x;
  if (t >= 2 * 192 * HW) return;
  int p = t % HW;
  int o = (t / HW) % 192;
  int b = t / (HW * 192);
  const float* fb = f + (size_t)b * 64 * HW + p;
  const float* wr = wqkv + o * 64;
  float acc = 0.f;
#pragma unroll 4
  for (int c = 0; c < 64; ++c) acc += wr[c] * fb[(size_t)c * HW];
  if (o < 64) {
    q[((size_t)b * 64 + o) * HW + p] = acc;
  } else if (o < 128) {
    k[((size_t)b * 64 + (o - 64)) * HW + p] = acc;
  } else {
    int oc = o - 128;                        // v channel = h*8 + c
    vT[(((size_t)(b * 8 + (oc >> 3))) * HW + p) * 8 + (oc & 7)] = acc;
  }
}

// ---------------- positional embedding table ----------------
__global__ void k_emb(const float* __restrict__ pos_h, const float* __restrict__ pos_w,
                      float* __restrict__ emb) {
  int t = blockIdx.x * blockDim.x + threadIdx.x;
  if (t >= 8 * HW) return;
  int c = t / HW, j = t % HW;
  emb[t] = pos_h[c * 48 + j / 48] + pos_w[c * 48 + j % 48];
}

// ---------------- 11x11 stride-8 conv + exact GELU ----------------
__global__ void k_conv(const float* __restrict__ q, const float* __restrict__ k,
                       const float* __restrict__ wq, const float* __restrict__ bq,
                       const float* __restrict__ wk, const float* __restrict__ bk,
                       float* __restrict__ qd, float* __restrict__ kd) {
  int t = blockIdx.x * blockDim.x + threadIdx.x;
  if (t >= 2 * 2 * 64 * 36) return;
  int which = t / (2 * 64 * 36);
  int idx = t % (2 * 64 * 36);
  int b = idx / (64 * 36);
  int oc = (idx / 36) % 64;
  int oo = idx % 36;
  int oy = oo / 6, ox = oo % 6;
  const float* src = which ? k : q;
  const float* wgt = which ? wk : wq;
  const float* bia = which ? bk : bq;
  float s = bia[oc];
  const float* sb = src + (size_t)b * 64 * HW;
  const float* wb = wgt + (size_t)oc * 64 * 121;
  for (int ic = 0; ic < 64; ++ic) {
    const float* sc = sb + (size_t)ic * HW;
    const float* wc = wb + ic * 121;
    for (int ky = 0; ky < 11; ++ky) {
      int iy = oy * 8 - 2 + ky;
      if (iy < 0 || iy >= 48) continue;
      for (int kx = 0; kx < 11; ++kx) {
        int ix = ox * 8 - 2 + kx;
        if (ix < 0 || ix >= 48) continue;
        s += wc[ky * 11 + kx] * sc[iy * 48 + ix];
      }
    }
  }
  s = 0.5f * s * (1.0f + erff(s * 0.70710678118654752f));  // exact GELU
  (which ? kd : qd)[idx] = s;
}

// ---------------- dots = (qd*SCALE)^T kd per (b,h), 36x36 ----------------
__global__ void k_dots(const float* __restrict__ qd, const float* __restrict__ kd,
                       float* __restrict__ dots) {
  int t = blockIdx.x * blockDim.x + threadIdx.x;
  if (t >= 16 * 36 * 36) return;
  int j = t % 36, i = (t / 36) % 36, bh = t / 1296;
  int b = bh >> 3, h = bh & 7;
  float s = 0.f;
#pragma unroll
  for (int c = 0; c < 8; ++c) {
    float a = qd[((b * 64) + h * 8 + c) * 36 + i] * SCALEF;
    float bb = kd[((b * 64) + h * 8 + c) * 36 + j];
    s += a * bb;
  }
  dots[t] = s;
}

// ---------------- fused flash-attention with WMMA ----------------
// one wave32 per (b, h, 16-row tile); never materializes the 2304x2304 matrix
__global__ __launch_bounds__(32) void k_attn(const float* __restrict__ q,
                                             const float* __restrict__ emb,
                                             const float* __restrict__ dots,
                                             const float* __restrict__ vT,
                                             float* __restrict__ out) {
  __shared__ float ptile[16 * 17];  // padded 16x16 P-tile for the A-operand bounce
  int bid = blockIdx.x;             // 0..2303
  int it = bid % NTILE;
  int bh = bid / NTILE;             // 0..15
  int b = bh >> 3, h = bh & 7;
  int lane = threadIdx.x & 31;
  int half = lane >> 4;
  int lm = lane & 15;
  int kb = half * 2;                // K pair handled by this lane half

  // A operand: pos-logit Q tile, rows = i, K = head dim (8 = 2 WMMAs of K=4)
  const float* qbase = q + ((size_t)(b * 64 + h * 8)) * HW + it * 16;
  v2f aq1, aq2;
  aq1.x = qbase[(size_t)(kb + 0) * HW + lm];
  aq1.y = qbase[(size_t)(kb + 1) * HW + lm];
  aq2.x = qbase[(size_t)(kb + 4) * HW + lm];
  aq2.y = qbase[(size_t)(kb + 5) * HW + lm];

  const float* drow = dots + ((size_t)bh * 36 + (it >> 2)) * 36;  // i-block fixed per tile
  const float* vtb = vT + (size_t)bh * HW * 8;

  // ---- pass 1: row maxima ----
  float pm[8];
#pragma unroll
  for (int r = 0; r < 8; ++r) pm[r] = -INFINITY;

  for (int jt = 0; jt < NTILE; ++jt) {
    int jc = jt * 16 + lm;
    v2f be1, be2;
    be1.x = emb[(size_t)(kb + 0) * HW + jc];
    be1.y = emb[(size_t)(kb + 1) * HW + jc];
    be2.x = emb[(size_t)(kb + 4) * HW + jc];
    be2.y = emb[(size_t)(kb + 5) * HW + jc];
    v8f acc = {};
    acc = wmma4(aq1, be1, acc);
    acc = wmma4(aq2, be2, acc);
    float s = drow[jt >> 2];  // upsampled dots: one scalar per 16x16 tile
#pragma unroll
    for (int r = 0; r < 8; ++r) pm[r] = fmaxf(pm[r], acc[r] + s);
  }
#pragma unroll
  for (int r = 0; r < 8; ++r) {  // reduce within each 16-lane half (wave32)
    float v = pm[r];
    v = fmaxf(v, __shfl_xor(v, 1, 32));
    v = fmaxf(v, __shfl_xor(v, 2, 32));
    v = fmaxf(v, __shfl_xor(v, 4, 32));
    v = fmaxf(v, __shfl_xor(v, 8, 32));
    pm[r] = v;
  }

  // ---- pass 2: exp, row sums, and P @ V^T via WMMA ----
  float ps[8];
#pragma unroll
  for (int r = 0; r < 8; ++r) ps[r] = 0.f;
  v8f oacc = {};

  for (int jt = 0; jt < NTILE; ++jt) {
    int jc = jt * 16 + lm;
    v2f be1, be2;
    be1.x = emb[(size_t)(kb + 0) * HW + jc];
    be1.y = emb[(size_t)(kb + 1) * HW + jc];
    be2.x = emb[(size_t)(kb + 4) * HW + jc];
    be2.y = emb[(size_t)(kb + 5) * HW + jc];
    v8f acc = {};
    acc = wmma4(aq1, be1, acc);
    acc = wmma4(aq2, be2, acc);
    float s = drow[jt >> 2];
#pragma unroll
    for (int r = 0; r < 8; ++r) {
      float e = expf(acc[r] + s - pm[r]);
      ps[r] += e;
      ptile[(r + 8 * half) * 17 + lm] = e;  // D layout -> LDS (same-wave, in-order)
    }
#pragma unroll
    for (int kc = 0; kc < 4; ++kc) {  // K=16 of P consumed as 4 WMMAs of K=4
      int x = kc * 4 + kb;
      v2f a, bb;
      a.x = ptile[lm * 17 + x];
      a.y = ptile[lm * 17 + x + 1];
      int j = jt * 16 + x;
      bb.x = (lm < 8) ? vtb[(size_t)(j + 0) * 8 + lm] : 0.f;  // N: 8 of 16 used
      bb.y = (lm < 8) ? vtb[(size_t)(j + 1) * 8 + lm] : 0.f;
      oacc = wmma4(a, bb, oacc);
    }
  }

#pragma unroll
  for (int r = 0; r < 8; ++r) {
    float v = ps[r];
    v += __shfl_xor(v, 1, 32);
    v += __shfl_xor(v, 2, 32);
    v += __shfl_xor(v, 4, 32);
    v += __shfl_xor(v, 8, 32);
    ps[r] = 1.0f / v;  // softmax denominator for row r + 8*half
  }
  if (lm < 8) {
    float* ob = out + ((size_t)(b * 64 + h * 8 + lm)) * HW + it * 16;
#pragma unroll
    for (int r = 0; r < 8; ++r) ob[r + 8 * half] = oacc[r] * ps[r];
  }
}

extern "C" void kernel_launch(void* const* d_in, const int* in_sizes, int n_in,
                              void* d_out, int out_size, void* d_ws, size_t ws_size,
                              hipStream_t stream) {
  (void)in_sizes; (void)n_in; (void)out_size; (void)ws_size;
  const float* f     = (const float*)d_in[0];
  const float* w_qkv = (const float*)d_in[1];
  const float* wq    = (const float*)d_in[2];
  const float* bq    = (const float*)d_in[3];
  const float* wk    = (const float*)d_in[4];
  const float* bk    = (const float*)d_in[5];
  const float* pos_h = (const float*)d_in[6];
  const float* pos_w = (const float*)d_in[7];
  float* out = (float*)d_out;
  float* ws = (float*)d_ws;

  float* q    = ws;            // 294912
  float* k    = ws + 294912;   // 294912
  float* vT   = ws + 589824;   // 294912  (vT[bh][j][c])
  float* qd   = ws + 884736;   // 4608
  float* kd   = ws + 889344;   // 4608
  float* dots = ws + 893952;   // 20736
  float* emb  = ws + 914688;   // 18432   (total ~3.7 MB)

  k_qkv <<<(2 * 192 * HW + 255) / 256, 256, 0, stream>>>(f, w_qkv, q, k, vT);
  k_emb <<<(8 * HW + 255) / 256, 256, 0, stream>>>(pos_h, pos_w, emb);
  k_conv<<<(2 * 2 * 64 * 36 + 255) / 256, 256, 0, stream>>>(q, k, wq, bq, wk, bk, qd, kd);
  k_dots<<<(16 * 36 * 36 + 255) / 256, 256, 0, stream>>>(qd, kd, dots);
  k_attn<<<2304, 32, 0, stream>>>(q, emb, dots, vT, out);
}